// HopfieldNetwork_2095944041023
// MI455X (gfx1250) — compile-verified
//
#include <hip/hip_runtime.h>
#include <hip/hip_bf16.h>

#define NN 4096

typedef __attribute__((ext_vector_type(2))) float v2f;
typedef __attribute__((ext_vector_type(4))) float v4f;
typedef __attribute__((ext_vector_type(8))) float v8f;

// ---------------------------------------------------------------------------
// Kernel 1: tiled transpose  WT[j][i] = W[i][j]   (coalesced read + write)
// ---------------------------------------------------------------------------
__global__ void hop_transpose(const float* __restrict__ W,
                              float* __restrict__ WT, int n) {
    __shared__ float tile[32][33];
    int bx = blockIdx.x * 32;   // source col block
    int by = blockIdx.y * 32;   // source row block
    int tx = threadIdx.x, ty = threadIdx.y;
    tile[ty][tx] = W[(size_t)(by + ty) * n + (bx + tx)];
    __syncthreads();
    WT[(size_t)(bx + ty) * n + (by + tx)] = tile[tx][ty];
}

// ---------------------------------------------------------------------------
// Kernel 2: a = W @ x  via V_WMMA_F32_16X16X4_F32.
// One wave (32 threads) per 16 output rows; K-loop of N/4 WMMAs.
// A fragment (16x4 f32, 2 VGPRs): lane l -> row M=l%16, K pair = 2*(l/16).
// B fragment built with all 16 columns identical (= state slice), so every
// column of C equals W_rows . x and the N-lane mapping is irrelevant.
// C (16x16 f32, 8 VGPRs): VGPR r holds M=r (lanes 0-15) / M=8+r (lanes 16-31).
// Fragment halves are contiguous+8B-aligned -> single v2f (b64) loads.
// ---------------------------------------------------------------------------
__global__ void hop_gemv_wmma(const float* __restrict__ W,
                              const float* __restrict__ x,
                              float* __restrict__ a, int n) {
    __shared__ float sx[NN];
    const int lane = threadIdx.x;          // blockDim.x == 32 (one wave)
    for (int k = lane; k < n; k += 32) sx[k] = x[k];
    __syncthreads();

    const int row0  = blockIdx.x * 16;
    const int m     = lane & 15;
    const int khalf = (lane >> 4) * 2;     // 0 or 2

    v8f c = {};
    const float* wrow = W + (size_t)(row0 + m) * n;
#pragma unroll 4
    for (int kb = 0; kb < n; kb += 4) {
        const int kc = kb + khalf;
        const v2f afrag = *reinterpret_cast<const v2f*>(wrow + kc);  // b64
        const v2f bfrag = *reinterpret_cast<const v2f*>(&sx[kc]);    // ds b64
        c = __builtin_amdgcn_wmma_f32_16x16x4_f32(
                /*neg_a=*/false, afrag, /*neg_b=*/false, bfrag,
                /*c_mod=*/(short)0, c, /*reuse_a=*/false, /*reuse_b=*/false);
    }
    // Column 0 of C lives in lane 0 (M=0..7) and lane 16 (M=8..15).
    if (lane == 0) {
#pragma unroll
        for (int r = 0; r < 8; ++r) a[row0 + r] = c[r];
    } else if (lane == 16) {
#pragma unroll
        for (int r = 0; r < 8; ++r) a[row0 + 8 + r] = c[r];
    }
}

// ---------------------------------------------------------------------------
// Kernel 3: asynchronous sweeps with incremental activation maintenance.
// Single workgroup, 1024 threads (32 waves). state/a live in LDS.
// Column i of W is addressed as Wc[i*s0 + k*s1]:
//   transposed path: (WT, s0=N, s1=1)  -> contiguous, b128 per thread
//   fallback path:   (W,  s0=1, s1=N)  -> strided scalar
// While processing column i we prefetch column i+1 (global_prefetch_b8),
// hiding the column fetch behind the serialized sign/barrier step.
// ---------------------------------------------------------------------------
__global__ void hop_sweep(const float* __restrict__ Wc, size_t s0, size_t s1,
                          const float* __restrict__ x,
                          const float* __restrict__ a_in,
                          const int* __restrict__ max_it,
                          float* __restrict__ out, int n) {
    __shared__ float st[NN];
    __shared__ float av[NN];
    __shared__ float dsh;
    const int tid  = threadIdx.x;
    const int nthr = blockDim.x;

    for (int k = tid; k < n; k += nthr) { st[k] = x[k]; av[k] = a_in[k]; }
    __syncthreads();

    const bool vec4 = (s1 == 1) && (n == 4 * nthr);   // transposed fast path
    v4f* av4 = reinterpret_cast<v4f*>(av);

    const int sweeps = *max_it;
    for (int s = 0; s < sweeps; ++s) {
        for (int i = 0; i < n; ++i) {
            const float* col = Wc + (size_t)i * s0;
            if (vec4 && (i + 1 < n) && ((tid & 3) == 0)) {
                // one prefetch per 64B line of the next column
                __builtin_prefetch(col + s0 + 4 * (size_t)tid, 0, 3);
            }
            if (tid == 0) {
                float act = av[i];
                float sn  = (act > 0.0f) ? 1.0f : ((act < 0.0f) ? -1.0f : 0.0f);
                dsh = sn - st[i];
                st[i] = sn;
            }
            __syncthreads();
            const float d = dsh;
            if (d != 0.0f) {
                if (vec4) {
                    const v4f w4 = *reinterpret_cast<const v4f*>(col + 4 * (size_t)tid);
                    av4[tid] = av4[tid] + w4 * d;
                } else {
#pragma unroll
                    for (int k = tid; k < n; k += nthr)
                        av[k] += col[(size_t)k * s1] * d;
                }
            }
            __syncthreads();
        }
    }
    for (int k = tid; k < n; k += nthr) out[k] = st[k];
}

// ---------------------------------------------------------------------------
extern "C" void kernel_launch(void* const* d_in, const int* in_sizes, int n_in,
                              void* d_out, int out_size, void* d_ws, size_t ws_size,
                              hipStream_t stream) {
    const float* x      = (const float*)d_in[0];
    const float* W      = (const float*)d_in[1];
    const int*   maxit  = (const int*)d_in[2];
    float*       out    = (float*)d_out;
    const int n = in_sizes[0];                 // 4096

    float* ws = (float*)d_ws;
    const size_t needT = ((size_t)n * n + n) * sizeof(float);
    const bool haveT = (ws_size >= needT);

    float* WT = ws;
    float* a  = haveT ? ws + (size_t)n * n : ws;

    if (haveT) {
        dim3 tb(32, 32);
        dim3 tg(n / 32, n / 32);
        hop_transpose<<<tg, tb, 0, stream>>>(W, WT, n);
    }

    hop_gemv_wmma<<<n / 16, 32, 0, stream>>>(W, x, a, n);

    if (haveT)
        hop_sweep<<<1, 1024, 0, stream>>>(WT, (size_t)n, (size_t)1,
                                          x, a, maxit, out, n);
    else
        hop_sweep<<<1, 1024, 0, stream>>>(W, (size_t)1, (size_t)n,
                                          x, a, maxit, out, n);
}